// MixtureExpertsAvancado_34600256537396
// MI455X (gfx1250) — compile-verified
//
#include <hip/hip_runtime.h>
#include <hip/hip_bf16.h>
#include <math.h>

// ---------------- problem constants (from reference) ----------------
#define DMODEL 1024
#define DHID   4096      // 4*D
#define NEXP   8

// ---------------- tiling ----------------
#define TM 128           // tokens per block tile
#define TN 128           // output columns per block tile   (NT = DMODEL/TN = 8)
#define KB 32            // k1 staging chunk for GEMM1 (over DMODEL)
#define KC 32            // hidden-dim chunk: GEMM1 N-tile == GEMM2 K-tile
#define NITER1 (DMODEL / KB)   // 32 inner GEMM1 staging steps

typedef __attribute__((ext_vector_type(2))) float v2f;
typedef __attribute__((ext_vector_type(8))) float v8f;
typedef __attribute__((ext_vector_type(4))) unsigned int u32x4;
typedef __attribute__((ext_vector_type(8))) int i32x8;
typedef __attribute__((ext_vector_type(4))) int i32x4;

#if __has_builtin(__builtin_amdgcn_tensor_load_to_lds)
#define HAVE_TDM 1
#else
#define HAVE_TDM 0
#endif

static __device__ __forceinline__ v8f wmma_f32(v2f a, v2f b, v8f c) {
  // V_WMMA_F32_16X16X4_F32 : D = A(16x4) x B(4x16) + C(16x16), wave32
  return __builtin_amdgcn_wmma_f32_16x16x4_f32(
      /*neg_a=*/false, a, /*neg_b=*/false, b,
      /*c_mod=*/(short)0, c, /*reuse_a=*/false, /*reuse_b=*/false);
}

#if HAVE_TDM
// 2D tile DMA: global (tile_y rows x tile_x 4-byte elems, row stride in elems)
// -> contiguous LDS at lds_byte_addr. Descriptor per CDNA5 ISA.
static __device__ __forceinline__ void tdm_load_2d(unsigned int lds_byte_addr,
                                                   const float* gaddr,
                                                   unsigned int tile_x,
                                                   unsigned int tile_y,
                                                   unsigned int row_stride_elems) {
  unsigned long long ga = (unsigned long long)(size_t)gaddr;
  u32x4 g0;
  g0.x = 1u;                                                  // count=1, user D#
  g0.y = lds_byte_addr;                                       // bits 63:32
  g0.z = (unsigned int)(ga & 0xFFFFFFFFull);                  // global_addr[31:0]
  g0.w = (unsigned int)((ga >> 32) & 0x01FFFFFFull)           // global_addr[56:32]
       | (2u << 30);                                          // type = 2 ("image")
  i32x8 g1;
  g1[0] = (int)(2u << 16);                                    // mask=0, data_size=4B
  g1[1] = (int)((tile_x & 0xFFFFu) << 16);                    // tensor_dim0[15:0]
  g1[2] = (int)(((tile_x >> 16) & 0xFFFFu) |                  // tensor_dim0[31:16]
                ((tile_y & 0xFFFFu) << 16));                  // tensor_dim1[15:0]
  g1[3] = (int)(((tile_y >> 16) & 0xFFFFu) |                  // tensor_dim1[31:16]
                ((tile_x & 0xFFFFu) << 16));                  // tile_dim0
  g1[4] = (int)(tile_y & 0xFFFFu);                            // tile_dim1 (tile_dim2=0)
  g1[5] = (int)row_stride_elems;                              // tensor_dim0_stride[31:0]
  g1[6] = 0;                                                  // stride[47:32], dim1_stride lo
  g1[7] = 0;
  i32x4 z4 = {0, 0, 0, 0};                                    // groups 2/3: 2D tensor
  i32x8 z8 = {0, 0, 0, 0, 0, 0, 0, 0};                        // trailing arg (6-arg form)
  __builtin_amdgcn_tensor_load_to_lds(g0, g1, z4, z4, z8, 0);
}
#endif

// ---------------- kernel 0: zero the per-expert counters ----------------
__global__ void moe_init_kernel(int* __restrict__ cnt) {
  if (threadIdx.x < NEXP) cnt[threadIdx.x] = 0;
}

// ---------------- kernel 1: gate -> softmax -> top-1 -> bucket ----------------
__global__ void moe_route_kernel(const float* __restrict__ x,
                                 const float* __restrict__ gate_w,  // [D, NEXP]
                                 const float* __restrict__ gate_b,  // [NEXP]
                                 int* __restrict__ cnt,             // [NEXP]
                                 int* __restrict__ perm,            // [NEXP, T]
                                 float* __restrict__ wgt,           // [T]
                                 int T) {
  int t = blockIdx.x * blockDim.x + threadIdx.x;
  if (t >= T) return;

  float acc[NEXP];
#pragma unroll
  for (int e = 0; e < NEXP; ++e) acc[e] = gate_b[e];

  const float4* xr = (const float4*)(x + (size_t)t * DMODEL);
  for (int d4 = 0; d4 < DMODEL / 4; ++d4) {
    float4 xv = xr[d4];
    float xs[4] = {xv.x, xv.y, xv.z, xv.w};
#pragma unroll
    for (int j = 0; j < 4; ++j) {
      int d = d4 * 4 + j;
      float4 g0 = *(const float4*)(gate_w + (size_t)d * NEXP);
      float4 g1 = *(const float4*)(gate_w + (size_t)d * NEXP + 4);
      acc[0] += xs[j] * g0.x; acc[1] += xs[j] * g0.y;
      acc[2] += xs[j] * g0.z; acc[3] += xs[j] * g0.w;
      acc[4] += xs[j] * g1.x; acc[5] += xs[j] * g1.y;
      acc[6] += xs[j] * g1.z; acc[7] += xs[j] * g1.w;
    }
  }

  // argmax with first-index tie-break (matches jax.lax.top_k)
  int best = 0; float bm = acc[0];
#pragma unroll
  for (int e = 1; e < NEXP; ++e) { if (acc[e] > bm) { bm = acc[e]; best = e; } }

  // softmax prob of the winner = 1 / sum(exp(acc - max))
  float s = 0.f;
#pragma unroll
  for (int e = 0; e < NEXP; ++e) s += expf(acc[e] - bm);
  wgt[t] = 1.0f / s;

  int pos = atomicAdd(&cnt[best], 1);
  perm[(size_t)best * T + pos] = t;
}

// ---------------- kernel 2: fused grouped FFN with f32 WMMA + TDM ----------------
__global__ __launch_bounds__(256, 2)
void moe_ffn_kernel(const float* __restrict__ x,
                    const float* __restrict__ w1,   // [NEXP, D, 4D]
                    const float* __restrict__ b1,   // [NEXP, 4D]
                    const float* __restrict__ w2,   // [NEXP, 4D, D]
                    const float* __restrict__ b2,   // [NEXP, D]
                    const int* __restrict__ cnt,
                    const int* __restrict__ perm,
                    const float* __restrict__ wgt,
                    float* __restrict__ out,        // [T, D]
                    int T) {
  const int nt = blockIdx.x;   // output-column tile
  const int mt = blockIdx.y;   // token tile
  const int e  = blockIdx.z;   // expert
  const int ne = cnt[e];
  const int m0 = mt * TM;
  if (m0 >= ne) return;        // block-uniform early exit
  const int n0 = nt * TN;

  __shared__ float sX [TM][KB + 1];        // +1 pad: conflict-free A-frag reads
  __shared__ float sW1[2][KB][KC];         // ping-pong, TDM-filled, contiguous
  __shared__ float sHc[8][16][KC + 1];     // wave-private gelu(H) re-layout
  __shared__ float sW2[KC][TN];            // TDM-filled, contiguous
  __shared__ int   sTok[TM];

  const int tid  = threadIdx.x;
  const int lane = tid & 31;
  const int wave = tid >> 5;               // 0..7
  const int l16  = lane & 15;
  const int half = lane >> 4;              // 0/1

  if (tid < TM) {
    int r = m0 + tid;
    sTok[tid] = (r < ne) ? perm[(size_t)e * T + r] : -1;
  }
  __syncthreads();

  const float* w1e = w1 + (size_t)e * DMODEL * DHID;
  const float* w2e = w2 + (size_t)e * DHID * DMODEL;
  const float* b1e = b1 + (size_t)e * DHID;
  const float* b2e = b2 + (size_t)e * DMODEL;

#if HAVE_TDM
  // flat LDS aperture keeps the LDS offset in addr[31:0]
  const unsigned int ldsW1_0 = (unsigned int)(size_t)&sW1[0][0][0];
  const unsigned int ldsW1_1 = (unsigned int)(size_t)&sW1[1][0][0];
  const unsigned int ldsW2   = (unsigned int)(size_t)&sW2[0][0];
#endif

  // X staging role: thread covers row tid/2, 16 consecutive cols
  const int xrow = tid >> 1;
  const int xcol = (tid & 1) * 16;
  const int xtok = sTok[xrow];

  v8f c[TN / 16] = {};                     // 8 accum tiles = 16x128 per wave

  for (int kk = 0; kk < DHID; kk += KC) {
    // Issue the W2 chunk DMA first: it overlaps the i=0 X staging and
    // (TDM ops of one wave complete in order) is forced done by the first
    // s_wait_tensorcnt<=1 below.
#if HAVE_TDM
    if (wave == 0) {
      tdm_load_2d(ldsW2, w2e + (size_t)kk * DMODEL + n0, TN, KC, DMODEL);
      tdm_load_2d(ldsW1_0, w1e + kk, KC, KB, DHID);   // W1 chunk i=0 -> buf0
    }
#endif
    v8f h[KC / 16] = {};                   // 16xKC H-fragment accumulators

    // ---------------- GEMM1: H = X[TM x D] * W1[:, kk:kk+KC] ----------------
    for (int i = 0; i < NITER1; ++i) {
      const int k1 = i * KB;
      const int p  = i & 1;
      __syncthreads();                     // (A) prior readers of buffers done
      // stage X tile (token gather -> per-thread vector loads)
      if (xtok >= 0) {
        const float4* src = (const float4*)(x + (size_t)xtok * DMODEL + k1 + xcol);
#pragma unroll
        for (int q = 0; q < 4; ++q) {
          float4 v = src[q];
          sX[xrow][xcol + q * 4 + 0] = v.x;
          sX[xrow][xcol + q * 4 + 1] = v.y;
          sX[xrow][xcol + q * 4 + 2] = v.z;
          sX[xrow][xcol + q * 4 + 3] = v.w;
        }
      } else {
#pragma unroll
        for (int q = 0; q < 16; ++q) sX[xrow][xcol + q] = 0.f;
      }
#if HAVE_TDM
      if (wave == 0) {
        if (i + 1 < NITER1) {              // prefetch next W1 chunk into other buf
          tdm_load_2d(p ? ldsW1_0 : ldsW1_1,
                      w1e + (size_t)(k1 + KB) * DHID + kk, KC, KB, DHID);
          __builtin_amdgcn_s_wait_tensorcnt(1);   // chunk i (and W2) landed
        } else {
          __builtin_amdgcn_s_wait_tensorcnt(0);   // everything landed
        }
      }
#else
      {
        int f  = tid * 4;                  // KB*KC = 1024 floats, 4 per thread
        int kr = f >> 5;
        int cc = f & 31;
        float4 v = *(const float4*)(w1e + (size_t)(k1 + kr) * DHID + kk + cc);
        sW1[p][kr][cc + 0] = v.x; sW1[p][kr][cc + 1] = v.y;
        sW1[p][kr][cc + 2] = v.z; sW1[p][kr][cc + 3] = v.w;
      }
      {
        int f  = tid * 16;                 // KC*TN = 4096 floats, 16 per thread
        if (i == 0) {
          int kr = f >> 7;
          int cc = f & 127;
          const float4* src = (const float4*)(w2e + (size_t)(kk + kr) * DMODEL + n0 + cc);
#pragma unroll
          for (int q = 0; q < 4; ++q) {
            float4 v = src[q];
            sW2[kr][cc + q * 4 + 0] = v.x;
            sW2[kr][cc + q * 4 + 1] = v.y;
            sW2[kr][cc + q * 4 + 2] = v.z;
            sW2[kr][cc + q * 4 + 3] = v.w;
          }
        }
      }
#endif
      __syncthreads();                     // (B) sX + sW1[p] (+sW2) visible

      for (int k4 = 0; k4 < KB; k4 += 4) {
        v2f a;                             // A 16x4: lane<16 -> K{0,1}, lane>=16 -> K{2,3}
        a.x = sX[wave * 16 + l16][k4 + 2 * half + 0];
        a.y = sX[wave * 16 + l16][k4 + 2 * half + 1];
#pragma unroll
        for (int nf = 0; nf < KC / 16; ++nf) {
          v2f b;                           // B 4x16 (mirrored striping)
          b.x = sW1[p][k4 + 2 * half + 0][nf * 16 + l16];
          b.y = sW1[p][k4 + 2 * half + 1][nf * 16 + l16];
          h[nf] = wmma_f32(a, b, h[nf]);
        }
      }
    }

    // ---------- bias + exact GELU, C-layout -> A-layout via wave-private LDS ----------
#pragma unroll
    for (int nf = 0; nf < KC / 16; ++nf) {
#pragma unroll
      for (int r = 0; r < 8; ++r) {
        int M   = r + 8 * half;            // C layout: VGPR r holds rows r and r+8
        int col = kk + nf * 16 + l16;
        float v = h[nf][r] + b1e[col];
        float g = 0.5f * v * (1.0f + erff(v * 0.70710678118654752f));
        sHc[wave][M][nf * 16 + l16] = g;
      }
    }

    // ---------------- GEMM2: C += gelu(H) * W2[kk:kk+KC, n0:n0+TN] ----------------
    // sW2 completed before the final (B) barrier above; sHc is wave-private.
    for (int k4 = 0; k4 < KC; k4 += 4) {
      v2f a;
      a.x = sHc[wave][l16][k4 + 2 * half + 0];
      a.y = sHc[wave][l16][k4 + 2 * half + 1];
#pragma unroll
      for (int nf = 0; nf < TN / 16; ++nf) {
        v2f b;
        b.x = sW2[k4 + 2 * half + 0][nf * 16 + l16];
        b.y = sW2[k4 + 2 * half + 1][nf * 16 + l16];
        c[nf] = wmma_f32(a, b, c[nf]);
      }
    }
    __syncthreads();                       // protect sW2/sW1 before next chunk's DMA
  }

  // ---------------- epilogue: + b2, * gate prob, scatter ----------------
#pragma unroll
  for (int r = 0; r < 8; ++r) {
    int M   = r + 8 * half;
    int tok = sTok[wave * 16 + M];
    if (tok >= 0) {
      float wg = wgt[tok];
#pragma unroll
      for (int nf = 0; nf < TN / 16; ++nf) {
        int col = n0 + nf * 16 + l16;
        out[(size_t)tok * DMODEL + col] = wg * (c[nf][r] + b2e[col]);
      }
    }
  }
}

// ---------------- launcher ----------------
extern "C" void kernel_launch(void* const* d_in, const int* in_sizes, int n_in,
                              void* d_out, int out_size, void* d_ws, size_t ws_size,
                              hipStream_t stream) {
  const float* x      = (const float*)d_in[0];
  const float* gate_w = (const float*)d_in[1];
  const float* gate_b = (const float*)d_in[2];
  const float* w1     = (const float*)d_in[3];
  const float* b1     = (const float*)d_in[4];
  const float* w2     = (const float*)d_in[5];
  const float* b2     = (const float*)d_in[6];
  float* out          = (float*)d_out;

  const int T = in_sizes[0] / DMODEL;      // 8192

  // workspace layout: [cnt: NEXP ints | perm: NEXP*T ints | wgt: T floats]
  char* ws   = (char*)d_ws;
  int*   cnt  = (int*)ws;
  int*   perm = (int*)(ws + 256);
  float* wgt  = (float*)(ws + 256 + (size_t)NEXP * T * sizeof(int));

  moe_init_kernel<<<1, 64, 0, stream>>>(cnt);
  moe_route_kernel<<<(T + 255) / 256, 256, 0, stream>>>(x, gate_w, gate_b,
                                                        cnt, perm, wgt, T);

  dim3 grid(DMODEL / TN, (T + TM - 1) / TM, NEXP);   // 8 x 64 x 8; inactive tiles exit
  moe_ffn_kernel<<<grid, 256, 0, stream>>>(x, w1, b1, w2, b2,
                                           cnt, perm, wgt, out, T);
}